// GIN_35716948034103
// MI455X (gfx1250) — compile-verified
//
#include <hip/hip_runtime.h>

#define N_NODES 100000
#define N_EDGES 1600000
#define NC 32
#define N_MID 8
#define N_TILES (N_NODES / 16)   // 6250, exact

typedef float v2f __attribute__((ext_vector_type(2)));
typedef float v8f __attribute__((ext_vector_type(8)));

// Branchless SELU: clamp the exp argument so it is always evaluated safely,
// then select.  Lowers to v_min / v_exp / v_add / v_cndmask / v_mul with no
// EXEC-mask branching (the previous version emitted s_and_saveexec per lane).
__device__ __forceinline__ float selu_f(float x) {
    const float alpha = 1.6732632423543772f;
    const float scale = 1.0507009873554805f;
    float neg = alpha * (__expf(fminf(x, 0.0f)) - 1.0f);
    float r = x > 0.0f ? x : neg;
    return scale * r;
}

// ---------------- generic float4 copy (agg = h) ----------------
__global__ __launch_bounds__(256) void copy_f4(const float4* __restrict__ src,
                                               float4* __restrict__ dst, int n4) {
    int t = blockIdx.x * blockDim.x + threadIdx.x;
    if (t < n4) dst[t] = src[t];
}

// ---------------- block 0: scalar-feature edge scatter ----------------
__global__ __launch_bounds__(256) void edge_scalar(const int* __restrict__ ei,
                                                   const float* __restrict__ x,
                                                   float* __restrict__ agg0) {
    int e = blockIdx.x * blockDim.x + threadIdx.x;
    if (e >= N_EDGES) return;
    int src = ei[e];
    int dst = ei[N_EDGES + e];
    unsafeAtomicAdd(agg0 + dst, x[src]);
}

// ---------------- block 0 MLP: [1 -> 16 -> 32], SELU ----------------
__global__ __launch_bounds__(256) void mlp0(const float* __restrict__ agg0,
                                            float* __restrict__ h,
                                            const float* __restrict__ W1,
                                            const float* __restrict__ b1,
                                            const float* __restrict__ W2,
                                            const float* __restrict__ b2) {
    int n = blockIdx.x * blockDim.x + threadIdx.x;
    if (n >= N_NODES) return;
    float a = agg0[n];
    float t[16];
#pragma unroll
    for (int j = 0; j < 16; ++j) t[j] = selu_f(fmaf(a, W1[j], b1[j]));
#pragma unroll
    for (int c = 0; c < NC; c += 4) {
        float4 s = *(const float4*)(b2 + c);
#pragma unroll
        for (int j = 0; j < 16; ++j) {
            float4 w = *(const float4*)(W2 + j * NC + c);
            s.x = fmaf(t[j], w.x, s.x);
            s.y = fmaf(t[j], w.y, s.y);
            s.z = fmaf(t[j], w.z, s.z);
            s.w = fmaf(t[j], w.w, s.w);
        }
        *(float4*)(h + (size_t)n * NC + c) = s;
    }
}

// ---------------- 32-channel edge scatter: 8 lanes per edge ----------------
__global__ __launch_bounds__(256) void edge_vec(const int* __restrict__ ei,
                                                const float* __restrict__ h,
                                                float* __restrict__ agg) {
    int t = blockIdx.x * blockDim.x + threadIdx.x;
    if (t >= N_EDGES * 8) return;
    int e = t >> 3;
    int q = t & 7;
    int src = ei[e];
    int dst = ei[N_EDGES + e];
    float4 v = *(const float4*)(h + (size_t)src * NC + q * 4);
    float* p = agg + (size_t)dst * NC + q * 4;
    unsafeAtomicAdd(p + 0, v.x);
    unsafeAtomicAdd(p + 1, v.y);
    unsafeAtomicAdd(p + 2, v.z);
    unsafeAtomicAdd(p + 3, v.w);
}

// ---------------- mid-block MLP via f32 WMMA, fused residual ----------------
// One wave per 16-node tile.  Layer1: T = selu(A(16x32) @ W1(32x16) + b1)
// (8x wmma 16x16x4).  T goes through padded LDS to re-lay into A format.
// Layer2: D = T(16x16) @ W2(16x32) + b2 + h_old  (2 N-halves x 4 wmma).
__global__ __launch_bounds__(256) void mlp_mid_wmma(const float* __restrict__ agg,
                                                    float* __restrict__ h,
                                                    const float* __restrict__ W1,
                                                    const float* __restrict__ b1,
                                                    const float* __restrict__ W2,
                                                    const float* __restrict__ b2) {
    __shared__ float lds[8][16][17];   // [wave][M][N], padded
    const int lane = threadIdx.x & 31;
    const int wid  = threadIdx.x >> 5;
    const int tile = blockIdx.x * 8 + wid;
    const bool active = tile < N_TILES;
    const int n16 = lane & 15;       // col index / row M for A operand
    const int hi  = lane >> 4;       // K half select per A/B layout
    const int base = tile * 16;

    v8f c1 = {};
    if (active) {
        // Layer 1: A = agg tile rows, B = W1 chunks
#pragma unroll
        for (int k0 = 0; k0 < 8; ++k0) {
            int kb = 4 * k0 + 2 * hi;
            v2f a = *(const v2f*)(agg + (size_t)(base + n16) * NC + kb);
            v2f b;
            b.x = W1[kb * 16 + n16];
            b.y = W1[(kb + 1) * 16 + n16];
            c1 = __builtin_amdgcn_wmma_f32_16x16x4_f32(
                     false, a, false, b, (short)0, c1, false, false);
        }
        float bb1 = b1[n16];
#pragma unroll
        for (int r = 0; r < 8; ++r) {
            int M = r + 8 * hi;                 // D layout: vgpr r, hi-half -> M
            lds[wid][M][n16] = selu_f(c1[r] + bb1);
        }
    }
    __syncthreads();
    if (active) {
        v8f c2a = {}, c2b = {};
#pragma unroll
        for (int k0 = 0; k0 < 4; ++k0) {
            int kb = 4 * k0 + 2 * hi;
            v2f a;
            a.x = lds[wid][n16][kb];
            a.y = lds[wid][n16][kb + 1];
            v2f bA, bB;
            bA.x = W2[kb * NC + n16];
            bA.y = W2[(kb + 1) * NC + n16];
            bB.x = W2[kb * NC + 16 + n16];
            bB.y = W2[(kb + 1) * NC + 16 + n16];
            c2a = __builtin_amdgcn_wmma_f32_16x16x4_f32(
                      false, a, false, bA, (short)0, c2a, false, false);
            c2b = __builtin_amdgcn_wmma_f32_16x16x4_f32(
                      false, a, false, bB, (short)0, c2b, false, false);
        }
        float bb2a = b2[n16];
        float bb2b = b2[16 + n16];
#pragma unroll
        for (int r = 0; r < 8; ++r) {
            int node = base + r + 8 * hi;
            size_t off = (size_t)node * NC + n16;
            float oa = h[off];
            float ob = h[off + 16];
            h[off]      = c2a[r] + bb2a + oa;   // residual fused
            h[off + 16] = c2b[r] + bb2b + ob;
        }
    }
}

// ---------------- block 9: out = agg @ W_last ----------------
__global__ __launch_bounds__(256) void final_dot(const float* __restrict__ agg,
                                                 const float* __restrict__ Wl,
                                                 float* __restrict__ out) {
    int n = blockIdx.x * blockDim.x + threadIdx.x;
    if (n >= N_NODES) return;
    float s = 0.0f;
#pragma unroll
    for (int c = 0; c < NC; c += 4) {
        float4 v = *(const float4*)(agg + (size_t)n * NC + c);
        float4 w = *(const float4*)(Wl + c);
        s = fmaf(v.x, w.x, s);
        s = fmaf(v.y, w.y, s);
        s = fmaf(v.z, w.z, s);
        s = fmaf(v.w, w.w, s);
    }
    out[n] = s;
}

extern "C" void kernel_launch(void* const* d_in, const int* in_sizes, int n_in,
                              void* d_out, int out_size, void* d_ws, size_t ws_size,
                              hipStream_t stream) {
    const float* x      = (const float*)d_in[0];
    const int*   ei     = (const int*)d_in[1];
    const float* W1_0   = (const float*)d_in[2];
    const float* b1_0   = (const float*)d_in[3];
    const float* W2_0   = (const float*)d_in[4];
    const float* b2_0   = (const float*)d_in[5];
    const float* W1_mid = (const float*)d_in[6];   // [8][32][16]
    const float* b1_mid = (const float*)d_in[7];   // [8][16]
    const float* W2_mid = (const float*)d_in[8];   // [8][16][32]
    const float* b2_mid = (const float*)d_in[9];   // [8][32]
    const float* W_last = (const float*)d_in[10];  // [32]

    float* h    = (float*)d_ws;                          // 12.8 MB
    float* agg  = h + (size_t)N_NODES * NC;              // 12.8 MB
    float* agg0 = agg + (size_t)N_NODES * NC;            // 0.4 MB

    const int B = 256;
    const int g_nodes   = (N_NODES + B - 1) / B;
    const int g_edges   = (N_EDGES + B - 1) / B;
    const int g_edgevec = (N_EDGES * 8 + B - 1) / B;
    const int g_copy32  = (N_NODES * NC / 4 + B - 1) / B;
    const int g_copy1   = (N_NODES / 4 + B - 1) / B;
    const int g_tiles   = (N_TILES + 7) / 8;

    // ---- block 0 ----
    copy_f4<<<g_copy1, B, 0, stream>>>((const float4*)x, (float4*)agg0, N_NODES / 4);
    edge_scalar<<<g_edges, B, 0, stream>>>(ei, x, agg0);
    mlp0<<<g_nodes, B, 0, stream>>>(agg0, h, W1_0, b1_0, W2_0, b2_0);

    // ---- blocks 1..8 ----
    for (int i = 0; i < N_MID; ++i) {
        copy_f4<<<g_copy32, B, 0, stream>>>((const float4*)h, (float4*)agg,
                                            N_NODES * NC / 4);
        edge_vec<<<g_edgevec, B, 0, stream>>>(ei, h, agg);
        mlp_mid_wmma<<<g_tiles, B, 0, stream>>>(agg, h,
                                                W1_mid + (size_t)i * NC * 16,
                                                b1_mid + (size_t)i * 16,
                                                W2_mid + (size_t)i * 16 * NC,
                                                b2_mid + (size_t)i * NC);
    }

    // ---- block 9 ----
    copy_f4<<<g_copy32, B, 0, stream>>>((const float4*)h, (float4*)agg,
                                        N_NODES * NC / 4);
    edge_vec<<<g_edgevec, B, 0, stream>>>(ei, h, agg);
    final_dot<<<g_nodes, B, 0, stream>>>(agg, W_last, (float*)d_out);
}